// PDELoss_52510270161124
// MI455X (gfx1250) — compile-verified
//
#include <hip/hip_runtime.h>
#include <math.h>

typedef float v2f __attribute__((ext_vector_type(2)));
typedef float v8f __attribute__((ext_vector_type(8)));

#define NPTS   262144
#define HDIM   128
#define TILE_M 16
#define ASTR   130                 // padded LDS row stride (floats): even (b64 align) + bank spread
#define ACH    (TILE_M * ASTR)     // per-channel activation size (floats)
#define NCH    5                   // primal, dz, dzz, dr, drr

// Branch-free fast transcendentals built on v_exp_f32 / v_log_f32.
__device__ __forceinline__ float tanh_fast(float x) {
    // 1 - 2/(1+e^{2x}); e^{2x}->inf gives 1, ->0 gives -1 (IEEE-safe saturation)
    float e = __expf(2.f * x);
    return 1.f - 2.f / (e + 1.f);
}
__device__ __forceinline__ float sigmoid_fast(float x) {
    return 1.f / (1.f + __expf(-x));
}
__device__ __forceinline__ float softplus_fast(float x) {
    // stable: max(x,0) + log(1+e^{-|x|})
    return fmaxf(x, 0.f) + __logf(1.f + __expf(-fabsf(x)));
}

// ---------------------------------------------------------------------------
// Dense layer on a 16-point tile, 5 JVP channels, f32 WMMA 16x16x4.
// Each wave processes N-tiles [ntStart, ntEnd). in/out: LDS [ch][16][stride].
// ---------------------------------------------------------------------------
template <bool TANH>
__device__ __forceinline__ void dense_layer(
    const float* __restrict__ W, const float* __restrict__ bias,
    int wrows, int wcols, int K, int ntStart, int ntEnd,
    const float* __restrict__ inA,
    float* __restrict__ outA, int outRowStride, int outChStride,
    int lane)
{
    const int half = lane >> 4;   // 0: K pair {k,k+1}, 1: {k+2,k+3}
    const int m    = lane & 15;   // row (A) / col (B,D) within tile

    for (int nt = ntStart; nt < ntEnd; ++nt) {
        const int n0   = nt * 16;
        const int col  = n0 + m;
        const int colc = (col < wcols) ? col : (wcols - 1);   // clamp for padded head

        v8f acc[NCH];
        #pragma unroll
        for (int c = 0; c < NCH; ++c)
            #pragma unroll
            for (int i = 0; i < 8; ++i) acc[c][i] = 0.f;

        for (int k0 = 0; k0 < K; k0 += 4) {
            const int kA  = k0 + 2 * half;
            const int kc0 = (kA     < wrows) ? kA       : (wrows - 1);
            const int kc1 = (kA + 1 < wrows) ? (kA + 1) : (wrows - 1);

            // B fragment (4x16 of W), shared by all 5 channels
            v2f bf;
            bf.x = W[kc0 * wcols + colc];
            bf.y = W[kc1 * wcols + colc];

            #pragma unroll
            for (int c = 0; c < NCH; ++c) {
                const float* ap = inA + c * ACH + m * ASTR + kA;  // 8B aligned (kA even)
                v2f af;
                af.x = ap[0];
                af.y = ap[1];
                acc[c] = __builtin_amdgcn_wmma_f32_16x16x4_f32(
                    false, af, false, bf, (short)0, acc[c], false, false);
            }
        }

        // Epilogue: bias on primal only, tanh second-order JVP chain elementwise.
        const float bn = bias ? bias[colc] : 0.f;
        #pragma unroll
        for (int i = 0; i < 8; ++i) {
            const int row = i + 8 * half;
            float u   = acc[0][i] + bn;
            float uz  = acc[1][i], uzz = acc[2][i];
            float ur  = acc[3][i], urr = acc[4][i];
            float v, vz, vzz, vr, vrr;
            if (TANH) {
                v = tanh_fast(u);
                float s = 1.f - v * v;
                vz  = s * uz;  vzz = s * uzz - 2.f * v * vz * uz;
                vr  = s * ur;  vrr = s * urr - 2.f * v * vr * ur;
            } else {
                v = u; vz = uz; vzz = uzz; vr = ur; vrr = urr;
            }
            float* op = outA + row * outRowStride + col;
            op[0 * outChStride] = v;
            op[1 * outChStride] = vz;
            op[2 * outChStride] = vzz;
            op[3 * outChStride] = vr;
            op[4 * outChStride] = vrr;
        }
    }
}

// ---------------------------------------------------------------------------
// Main kernel: 4 waves (128 threads) cooperate on one 16-point tile.
// Each wave owns 2 of the 8 N-tiles per hidden layer; wave 0 does the head.
// ---------------------------------------------------------------------------
__global__ void __launch_bounds__(128)
pinn_soot_wmma_kernel(const float* __restrict__ inputs,
                      const int*   __restrict__ fuel_ids,
                      const float* __restrict__ W1, const float* __restrict__ b1,
                      const float* __restrict__ W2, const float* __restrict__ b2,
                      const float* __restrict__ W3, const float* __restrict__ b3,
                      const float* __restrict__ Wo, const float* __restrict__ bo,
                      const float* __restrict__ emb,
                      float* __restrict__ ws)
{
    __shared__ float sA[NCH * ACH];
    __shared__ float sB[NCH * ACH];
    __shared__ float sO[NCH * 16 * 16];   // head output staging [ch][pt][col]

    const int tid  = threadIdx.x;
    const int lane = tid & 31;
    const int wv   = tid >> 5;
    const int p0   = blockIdx.x * TILE_M;

    // ---- stage inputs: [x0..x3, emb0..7, 0,0,0,0]; tangent one-hots -------
    if (tid < 16) {
        const int p   = p0 + tid;
        const int fid = fuel_ids[p];
        #pragma unroll
        for (int j = 0; j < 16; ++j) {
            float x = 0.f;
            if (j < 4)       x = inputs[p * 4 + j];
            else if (j < 12) x = emb[fid * 8 + (j - 4)];
            sA[0 * ACH + tid * ASTR + j] = x;
            sA[1 * ACH + tid * ASTR + j] = (j == 1) ? 1.f : 0.f;  // Ez
            sA[2 * ACH + tid * ASTR + j] = 0.f;
            sA[3 * ACH + tid * ASTR + j] = (j == 2) ? 1.f : 0.f;  // Er
            sA[4 * ACH + tid * ASTR + j] = 0.f;
        }
    }
    __syncthreads();

    const int nt0 = wv * 2, nt1 = wv * 2 + 2;
    dense_layer<true >(W1, b1,   12, HDIM,   16, nt0, nt1, sA, sB, ASTR, ACH, lane);
    __syncthreads();
    dense_layer<true >(W2, b2, HDIM, HDIM, HDIM, nt0, nt1, sB, sA, ASTR, ACH, lane);
    __syncthreads();
    dense_layer<true >(W3, b3, HDIM, HDIM, HDIM, nt0, nt1, sA, sB, ASTR, ACH, lane);
    __syncthreads();
    if (wv == 0) {  // uniform per-wave branch; EXEC all-ones inside wave 0
        dense_layer<false>(Wo, bo, HDIM, 7, HDIM, 0, 1, sB, sO, 16, 256, lane);
    }
    __syncthreads();

    // ---- per-point PDE residuals (wave 0, lanes 0..15, one point each) -----
    if (wv == 0) {
        float accE = 0.f, accS = 0.f;
        if (lane < 16) {
            const int p = p0 + lane;
            #define OT(c, j) sO[(c) * 256 + lane * 16 + (j)]
            // Temperature head: T = 300 + 2000*sigmoid(o0)
            float o0 = OT(0,0), o0z = OT(1,0), o0zz = OT(2,0), o0r = OT(3,0), o0rr = OT(4,0);
            float sg  = sigmoid_fast(o0);
            float T   = 300.f + 2000.f * sg;
            float sg1 = sg * (1.f - sg);             // sigma'
            float sg2 = sg1 * (1.f - 2.f * sg);      // sigma''
            float Tz  = 2000.f * sg1 * o0z;
            float Tzz = 2000.f * (sg2 * o0z * o0z + sg1 * o0zz);
            float Tr  = 2000.f * sg1 * o0r;
            float Trr = 2000.f * (sg2 * o0r * o0r + sg1 * o0rr);
            // Soot volume fraction: fv = softplus(o1)*1e-6
            float o1 = OT(0,1), o1z = OT(1,1);
            float fv  = softplus_fast(o1) * 1e-6f;
            float fvz = sigmoid_fast(o1) * o1z * 1e-6f;
            // Softmax over o[2..6]; need Y[3] (prec) and Y[4] (O2)
            float mx = OT(0,2);
            #pragma unroll
            for (int j = 3; j < 7; ++j) mx = fmaxf(mx, OT(0,j));
            float Z = 0.f, e3 = 0.f, e4 = 0.f;
            #pragma unroll
            for (int j = 0; j < 5; ++j) {
                float e = __expf(OT(0, 2 + j) - mx);
                Z += e;
                if (j == 3) e3 = e;
                if (j == 4) e4 = e;
            }
            float Yp = e3 / Z, Yo2 = e4 / Z;
            // Residuals
            float r  = inputs[p * 4 + 2];
            float rs = fmaxf(r, 1e-6f);
            float energy = 180.f * Tz - 0.06f * (Trr + Tr / rs) - 0.06f * Tzz; // RHO*CP*VZ, KT
            float nuc  = 10000.f  * Yp  * __expf(-20000.f / T);
            float grow = 5000.f   * Yp  * fv * __expf(-12000.f / T);
            float ox   = 100000.f * Yo2 * fv * sqrtf(T) * __expf(-16000.f / T);
            float soot = 0.5f * fvz - nuc - grow + ox;   // VZ = 0.5
            accE = energy * energy;
            accS = soot * soot;
            #undef OT
        }
        // wave32 butterfly reduce (lanes 16..31 contribute zeros)
        #pragma unroll
        for (int off = 16; off >= 1; off >>= 1) {
            accE += __shfl_xor(accE, off, 32);
            accS += __shfl_xor(accS, off, 32);
        }
        if (lane == 0) {
            ws[2 * blockIdx.x + 0] = accE;
            ws[2 * blockIdx.x + 1] = accS;
        }
    }
}

// ---------------------------------------------------------------------------
// Deterministic final reduction: fixed strided order + fixed tree.
// ---------------------------------------------------------------------------
__global__ void __launch_bounds__(256)
pinn_soot_reduce_kernel(const float* __restrict__ ws, float* __restrict__ out,
                        int nblocks, float invN)
{
    __shared__ float se[256], ss[256];
    const int tid = threadIdx.x;
    float e = 0.f, s = 0.f;
    for (int i = tid; i < nblocks; i += 256) {
        e += ws[2 * i + 0];
        s += ws[2 * i + 1];
    }
    se[tid] = e; ss[tid] = s;
    __syncthreads();
    for (int st = 128; st > 0; st >>= 1) {
        if (tid < st) { se[tid] += se[tid + st]; ss[tid] += ss[tid + st]; }
        __syncthreads();
    }
    if (tid == 0) {
        float le = se[0] * invN, ls = ss[0] * invN;
        out[0] = le;
        out[1] = ls;
        out[2] = le + ls;
    }
}

extern "C" void kernel_launch(void* const* d_in, const int* in_sizes, int n_in,
                              void* d_out, int out_size, void* d_ws, size_t ws_size,
                              hipStream_t stream)
{
    const float* inputs   = (const float*)d_in[0];
    const int*   fuel_ids = (const int*)  d_in[1];
    const float* W1 = (const float*)d_in[2];
    const float* b1 = (const float*)d_in[3];
    const float* W2 = (const float*)d_in[4];
    const float* b2 = (const float*)d_in[5];
    const float* W3 = (const float*)d_in[6];
    const float* b3 = (const float*)d_in[7];
    const float* Wo = (const float*)d_in[8];
    const float* bo = (const float*)d_in[9];
    const float* emb = (const float*)d_in[10];
    float* out = (float*)d_out;
    float* ws  = (float*)d_ws;   // needs 2 * (N/16) floats = 128 KB

    const int nblocks = NPTS / TILE_M;  // 16384
    pinn_soot_wmma_kernel<<<nblocks, 128, 0, stream>>>(
        inputs, fuel_ids, W1, b1, W2, b2, W3, b3, Wo, bo, emb, ws);
    pinn_soot_reduce_kernel<<<1, 256, 0, stream>>>(ws, out, nblocks, 1.0f / (float)NPTS);
}